// FM_v_38560216383899
// MI455X (gfx1250) — compile-verified
//
#include <hip/hip_runtime.h>

typedef __attribute__((ext_vector_type(2))) float v2f;
typedef __attribute__((ext_vector_type(8))) float v8f;

#define Cn   8
#define Vn   12
#define Dn   64
#define Bn   32768
#define TPB  256
#define WAVES 8
#define ROWS_PER_BLOCK (WAVES * 16)        // 128 rows per block
#define NBLK (Bn / ROWS_PER_BLOCK)         // 256 blocks

__device__ __forceinline__ float softplus_f(float x) {
    // numerically stable log(1+exp(x))
    return fmaxf(x, 0.0f) + log1pf(expf(-fabsf(x)));
}

__global__ __launch_bounds__(TPB) void fm_main(
    const int*   __restrict__ idx,    // [C,B]
    const float* __restrict__ label,  // [B,2]
    const float* __restrict__ posw,   // [B,2]
    const float* __restrict__ meant,  // [C,V,D]
    const float* __restrict__ stdt,   // [C,V,D]
    const float* __restrict__ act,    // [2,D]
    const float* __restrict__ noise,  // [B*D]
    float*       __restrict__ out,    // [2B | 1 | B]
    float*       __restrict__ blkpart)// [NBLK]
{
    __shared__ float mu_s[Cn * Vn * Dn];    // 24 KB
    __shared__ float sp_s[Cn * Vn * Dn];    // 24 KB (softplus precomputed)
    __shared__ float act_s[16 * Dn];        // 4 KB, rows 2..15 zero-padded (B fragment source)
    __shared__ float red_s[WAVES];

    const int tid = threadIdx.x;
    for (int i = tid; i < Cn * Vn * Dn; i += TPB) {
        mu_s[i] = meant[i];
        sp_s[i] = softplus_f(stdt[i]);
    }
    for (int i = tid; i < 16 * Dn; i += TPB)
        act_s[i] = (i < 2 * Dn) ? act[i] : 0.0f;
    __syncthreads();

    const int lane = tid & 31;
    const int wv   = tid >> 5;
    const int m    = lane & 15;      // row within 16-row tile (also WMMA column id n)
    const int h    = lane >> 4;      // half-wave: selects K sub-pair / M-half of accum
    const int tile = blockIdx.x * WAVES + wv;
    const int b0   = tile * 16;
    const int b    = b0 + m;         // this lane computes s for row b, d = 4*kc + 2h + {0,1}

    int vi[Cn];
#pragma unroll
    for (int c = 0; c < Cn; ++c) vi[c] = idx[c * Bn + b];

    v8f acc = {};                    // 16x16 f32 accumulator; cols 0,1 = inferences
    float s2 = 0.0f, e2 = 0.0f;
    const float2* nz = (const float2*)(noise + (size_t)b * Dn);

#pragma unroll
    for (int kc = 0; kc < Dn / 4; ++kc) {
        const int d0 = 4 * kc + 2 * h;           // even -> 8B-aligned float2 load
        float2 w2 = nz[d0 >> 1];
        const float w0 = 0.01f * w2.x;
        const float w1 = 0.01f * w2.y;
        float s0 = 0.0f, s1 = 0.0f;
#pragma unroll
        for (int c = 0; c < Cn; ++c) {
            const int base = (c * Vn + vi[c]) * Dn + d0;
            const float e0 = fmaf(sp_s[base],     w0, mu_s[base]);
            const float e1 = fmaf(sp_s[base + 1], w1, mu_s[base + 1]);
            s0 += e0; s1 += e1;
            e2 = fmaf(e0, e0, e2);
            e2 = fmaf(e1, e1, e2);
        }
        s2 = fmaf(s0, s0, s2);
        s2 = fmaf(s1, s1, s2);
        // A fragment (16x4 f32): VGPR0 = A[m, 2h], VGPR1 = A[m, 2h+1]
        v2f a;  a.x = s0;                a.y = s1;
        // B fragment (4x16 f32): VGPR0 = B[2h, n=m], VGPR1 = B[2h+1, n=m]; B[k,n] = act[n][d]
        v2f bb; bb.x = act_s[m * Dn + d0]; bb.y = act_s[m * Dn + d0 + 1];
        acc = __builtin_amdgcn_wmma_f32_16x16x4_f32(
            /*neg_a=*/false, a, /*neg_b=*/false, bb,
            /*c_mod=*/(short)0, acc, /*reuse_a=*/false, /*reuse_b=*/false);
    }

    // pairwise: row sums are split across lane pair (L, L+16)
    const float s2f = s2 + __shfl_xor(s2, 16, 32);
    const float e2f = e2 + __shfl_xor(e2, 16, 32);
    if (h == 0)
        out[2 * Bn + 1 + b] = 0.5f * (s2f - e2f);

    // inferences (accumulator cols 0,1) + weighted-loss partial
    float lossp = 0.0f;
    if (m < 2) {
        const int n = m;
#pragma unroll
        for (int r = 0; r < 8; ++r) {
            const int br   = b0 + h * 8 + r;     // VGPR r holds M = h*8 + r
            const float iv = acc[r];
            out[br * 2 + n] = iv;
            const float dlt = iv - label[br * 2 + n];
            lossp = fmaf(posw[br * 2 + n] * dlt, dlt, lossp);
        }
    }
#pragma unroll
    for (int off = 16; off >= 1; off >>= 1)
        lossp += __shfl_xor(lossp, off, 32);
    if (lane == 0) red_s[wv] = lossp;
    __syncthreads();
    if (tid == 0) {
        float t = 0.0f;
#pragma unroll
        for (int w = 0; w < WAVES; ++w) t += red_s[w];
        blkpart[blockIdx.x] = t;                 // fixed-order -> deterministic
    }
}

__global__ __launch_bounds__(NBLK) void fm_reduce(
    const float* __restrict__ part, float* __restrict__ out)
{
    __shared__ float s[NBLK];
    const int t = threadIdx.x;
    s[t] = part[t];
    __syncthreads();
    for (int off = NBLK / 2; off > 0; off >>= 1) {
        if (t < off) s[t] += s[t + off];
        __syncthreads();
    }
    if (t == 0) out[2 * Bn] = s[0] / (float)Bn;  // overwrite each call (replay-safe)
}

extern "C" void kernel_launch(void* const* d_in, const int* in_sizes, int n_in,
                              void* d_out, int out_size, void* d_ws, size_t ws_size,
                              hipStream_t stream) {
    const int*   idx   = (const int*)d_in[0];
    const float* label = (const float*)d_in[1];
    const float* posw  = (const float*)d_in[2];
    const float* meant = (const float*)d_in[3];
    const float* stdt  = (const float*)d_in[4];
    const float* act   = (const float*)d_in[5];
    const float* noise = (const float*)d_in[6];
    float* out  = (float*)d_out;
    float* part = (float*)d_ws;   // needs NBLK*4 = 1 KB

    fm_main<<<NBLK, TPB, 0, stream>>>(idx, label, posw, meant, stdt, act, noise, out, part);
    fm_reduce<<<1, NBLK, 0, stream>>>(part, out);
}